// RandomDrop_49263274885891
// MI455X (gfx1250) — compile-verified
//
#include <hip/hip_runtime.h>

#define NB   32      // number of batch segments
#define BINS 4096    // score histogram bins per segment
#define CAP  2048    // max threshold-bin points collected per segment

typedef __attribute__((ext_vector_type(2))) float v2f;
typedef __attribute__((ext_vector_type(8))) float v8f;

__device__ __forceinline__ int bin_of(float s) {
  int b = (int)(s * (float)BINS);
  if (b < 0) b = 0;
  if (b > BINS - 1) b = BINS - 1;
  return b;
}

// ---------------------------------------------------------------- zero hist
__global__ void zero_kernel(unsigned* __restrict__ p, int n) {
  int i = blockIdx.x * blockDim.x + threadIdx.x;
  if (i < n) p[i] = 0u;
}

// ------------------------------------------------------- per-segment histogram
__global__ void hist_kernel(const int* __restrict__ bidx,
                            const float* __restrict__ scores,
                            int n, unsigned* __restrict__ hist) {
  int i = blockIdx.x * blockDim.x + threadIdx.x;
  if (i >= n) return;
  int b = bidx[i];
  int bin = bin_of(scores[i]);
  atomicAdd(&hist[b * BINS + bin], 1u);
}

// --------------------------------------- segment starts + threshold bin search
// seginfo[b] = {start, count, t, r}
__global__ void thresh_kernel(const unsigned* __restrict__ hist,
                              int* __restrict__ seginfo) {
  __shared__ unsigned scnt[NB];
  int b = threadIdx.x;               // one thread per segment
  if (b >= NB) return;
  unsigned cnt = 0;
  for (int j = 0; j < BINS; ++j) cnt += hist[b * BINS + j];
  scnt[b] = cnt;
  __syncthreads();
  unsigned start = 0;
  for (int j = 0; j < b; ++j) start += scnt[j];   // exclusive prefix (32 elems)
  unsigned ndrop = cnt >> 1;                      // floor(n_b * 0.5)
  unsigned acc = 0, r = 0;
  int t = BINS - 1;
  for (int j = 0; j < BINS; ++j) {
    unsigned h = hist[b * BINS + j];
    if (acc + h >= ndrop) { t = j; r = ndrop - acc; break; }
    acc += h;
  }
  seginfo[b * 4 + 0] = (int)start;
  seginfo[b * 4 + 1] = (int)cnt;
  seginfo[b * 4 + 2] = t;
  seginfo[b * 4 + 3] = (int)r;
}

// ------------------------------------------------- mask: one block per segment
__global__ __launch_bounds__(1024)
void mask_kernel(const float* __restrict__ scores,
                 const int* __restrict__ seginfo,
                 float* __restrict__ mask) {
  int b     = blockIdx.x;
  int start = seginfo[b * 4 + 0];
  int cnt   = seginfo[b * 4 + 1];
  int t     = seginfo[b * 4 + 2];
  int r     = seginfo[b * 4 + 3];

  __shared__ int   npend;
  __shared__ float pscore[CAP];
  __shared__ int   pidx[CAP];
  if (threadIdx.x == 0) npend = 0;
  __syncthreads();

  for (int off = threadIdx.x; off < cnt; off += blockDim.x) {
    int i = start + off;
    float s = scores[i];
    int bin = bin_of(s);
    if (bin > t)      mask[i] = 1.0f;
    else if (bin < t) mask[i] = 0.0f;
    else {
      int p = atomicAdd(&npend, 1);
      if (p < CAP) { pscore[p] = s; pidx[p] = i; }
    }
  }
  __syncthreads();

  int m = npend < CAP ? npend : CAP;
  // exact rank among threshold-bin points; stable tie-break (score, index)
  for (int e = threadIdx.x; e < m; e += blockDim.x) {
    float s = pscore[e];
    int   i = pidx[e];
    int rank = 0;
    for (int q = 0; q < m; ++q) {
      float sq = pscore[q];
      if (sq < s || (sq == s && pidx[q] < i)) rank++;
    }
    mask[i] = (rank >= r) ? 1.0f : 0.0f;
  }
}

// ---------------------------------------------- apply: D = diag(mask16) x F16x64
// via V_WMMA_F32_16X16X4_F32 (bit-exact since mask is 0/1). One wave per
// 16-row x 64-col tile. Single-touch feats/out traffic is marked non-temporal
// so the 512MB stream doesn't evict the reusable mask/scores from L2.
__global__ __launch_bounds__(256)
void apply_kernel(const float* __restrict__ feats,
                  const float* __restrict__ mask,
                  float* __restrict__ out, int nrows) {
  const int lane   = threadIdx.x & 31;
  const int wave   = threadIdx.x >> 5;
  const int lane16 = lane & 15;
  const int hi     = lane >> 4;                    // 0: K=0,1 half; 1: K=2,3 half
  const long tile  = (long)blockIdx.x * 8 + wave;  // 16-row tile
  const long row0  = tile * 16;
  if (row0 >= nrows) return;                       // uniform per wave

  // prefetch next block's feats tile (global_prefetch_b8)
  {
    long pr = row0 + 128;
    if (pr < nrows)
      __builtin_prefetch(&feats[pr * 64 + (long)lane * 4], 0, 0);
  }

  const bool full = (row0 + 16 <= (long)nrows);    // uniform per wave

  // A = diag(mask) split into four 16x4 slices (columns 4k..4k+3).
  // A-layout (32-bit, 16x4): lanes 0-15 hold K = {0,1}, lanes 16-31 K = {2,3}.
  long mrow = row0 + lane16;
  if (mrow > (long)nrows - 1) mrow = nrows - 1;
  float mval = mask[mrow];                          // RT: reused, L2-resident

  v2f a[4];
#pragma unroll
  for (int k = 0; k < 4; ++k) {
    int j0 = 4 * k + (hi ? 2 : 0);
    int j1 = 4 * k + (hi ? 3 : 1);
    a[k].x = (lane16 == j0) ? mval : 0.0f;
    a[k].y = (lane16 == j1) ? mval : 0.0f;
  }

  if (full) {
    // One base address per wave; all 32 loads + 32 stores use constant offsets.
    const float* __restrict__ fbase =
        feats + (row0 + (hi ? 2 : 0)) * 64 + lane16;   // B slice rows j0, j0+1
    float* __restrict__ obase =
        out + (row0 + hi * 8) * 64 + lane16;           // D rows v + hi*8
#pragma unroll
    for (int ct = 0; ct < 4; ++ct) {                   // four 16-col tiles
      const int c0 = ct * 16;
      v8f acc = {};
#pragma unroll
      for (int k = 0; k < 4; ++k) {
        v2f bm;
        bm.x = __builtin_nontemporal_load(&fbase[k * 256 + c0]);      // row j0
        bm.y = __builtin_nontemporal_load(&fbase[k * 256 + 64 + c0]); // row j0+1
        acc = __builtin_amdgcn_wmma_f32_16x16x4_f32(
            false, a[k], false, bm, (short)0, acc, false, false);
      }
#pragma unroll
      for (int v = 0; v < 8; ++v)
        __builtin_nontemporal_store(acc[v], &obase[v * 64 + c0]);
    }
  } else {
    // Rare tail tile: clamp loads, predicate stores (WMMA still full-EXEC).
#pragma unroll
    for (int ct = 0; ct < 4; ++ct) {
      const int c0 = ct * 16;
      v8f acc = {};
#pragma unroll
      for (int k = 0; k < 4; ++k) {
        long r0 = row0 + 4 * k + (hi ? 2 : 0);
        long r1 = row0 + 4 * k + (hi ? 3 : 1);
        if (r0 > (long)nrows - 1) r0 = nrows - 1;
        if (r1 > (long)nrows - 1) r1 = nrows - 1;
        v2f bm;
        bm.x = feats[r0 * 64 + c0 + lane16];
        bm.y = feats[r1 * 64 + c0 + lane16];
        acc = __builtin_amdgcn_wmma_f32_16x16x4_f32(
            false, a[k], false, bm, (short)0, acc, false, false);
      }
#pragma unroll
      for (int v = 0; v < 8; ++v) {
        long orow = row0 + v + hi * 8;
        if (orow < nrows)
          out[orow * 64 + c0 + lane16] = acc[v];
      }
    }
  }
}

extern "C" void kernel_launch(void* const* d_in, const int* in_sizes, int n_in,
                              void* d_out, int out_size, void* d_ws, size_t ws_size,
                              hipStream_t stream) {
  const float* feats  = (const float*)d_in[0];
  const int*   bidx   = (const int*)d_in[1];
  const float* scores = (const float*)d_in[2];
  float*       out    = (float*)d_out;
  const int n = in_sizes[1];          // number of points (1M)

  // workspace layout: [ mask: n floats ][ hist: 32*4096 u32 ][ seginfo: 32*4 i32 ]
  char* ws = (char*)d_ws;
  float*    mask    = (float*)ws;
  unsigned* hist    = (unsigned*)(ws + (size_t)n * sizeof(float));
  int*      seginfo = (int*)(ws + (size_t)n * sizeof(float)
                                + (size_t)NB * BINS * sizeof(unsigned));

  zero_kernel<<<(NB * BINS + 255) / 256, 256, 0, stream>>>(hist, NB * BINS);
  hist_kernel<<<(n + 255) / 256, 256, 0, stream>>>(bidx, scores, n, hist);
  thresh_kernel<<<1, NB, 0, stream>>>(hist, seginfo);
  mask_kernel<<<NB, 1024, 0, stream>>>(scores, seginfo, mask);

  const long ntiles  = ((long)n + 15) / 16;
  const int  nblocks = (int)((ntiles + 7) / 8);   // 8 waves (tiles) per block
  apply_kernel<<<nblocks, 256, 0, stream>>>(feats, mask, out, n);
}